// LinearPredictiveCoefficientsToParcorCoefficients_47502338294559
// MI455X (gfx1250) — compile-verified
//
#include <hip/hip_runtime.h>
#include <stdint.h>

// LPC -> PARCOR (reverse Levinson-Durbin step-down), 2048x1024 rows of 33 f32.
// Memory-bound floor ~24 us (554 MB @ 23.3 TB/s); VALU cost ~46 instr/row is
// the same order, so we (a) DMA the 132-byte rows as 256x33 f32 2D tiles via
// the Tensor Data Mover (fully coalesced, zero VGPR address pressure),
// (b) run the recursion fully unrolled in VGPRs, and (c) use v_rcp_f32
// (TRANS op, co-executes with VALU) instead of the full-precision divide
// expansion. LDS row stride = 33 dwords, gcd(33,64)=1 => bank-conflict-free.

#define GAMMA 1.0f
#define ROWS_PER_BLOCK 256
#define ORDER 32   // M: number of reflection coefficients
#define W 33       // row width: gain + 32 LPC coefficients

typedef uint32_t u32x4 __attribute__((ext_vector_type(4)));
typedef int32_t  i32x8 __attribute__((ext_vector_type(8)));
typedef int32_t  i32x4 __attribute__((ext_vector_type(4)));

// Build the Tensor DMA Descriptor (D#) for a 2D tile of `nrows` rows x 33 f32
// starting at `gaddr` (tile start, row-major, row stride 33 elements) and
// issue TENSOR_LOAD_TO_LDS / TENSOR_STORE_FROM_LDS (6-arg builtin on this
// toolchain). 2D tensor => groups 2/3 and the trailing group are inert.
__device__ static inline void tdm_issue(bool is_store, uint32_t lds_byte_off,
                                        uint64_t gaddr, uint32_t nrows) {
    // ---- D# group 0 (128b): count / lds_addr / global_addr / type ----
    u32x4 g0;
    g0[0] = 1u;                                     // count=1 (valid), user mode
    g0[1] = lds_byte_off;                           // lds_addr[31:0]
    g0[2] = (uint32_t)(gaddr & 0xFFFFFFFFu);        // global_addr[31:0]
    g0[3] = (uint32_t)((gaddr >> 32) & 0x01FFFFFFu) // global_addr[56:32]
          | (2u << 30);                             // type=2 ("image") @ 127:126
    // ---- D# group 1 (256b): geometry ----
    i32x8 g1;
    g1[0] = (int32_t)(2u << 16);                    // workgroup_mask=0, data_size=2 (4B)
    g1[1] = (int32_t)((uint32_t)W << 16);           // tensor_dim0[15:0]=33 @ 63:48
    g1[2] = (int32_t)((nrows & 0xFFFFu) << 16);     // tensor_dim0 hi=0, tensor_dim1 lo
    g1[3] = (int32_t)(((nrows >> 16) & 0xFFFFu)     // tensor_dim1 hi
          | ((uint32_t)W << 16));                   // tile_dim0=33 @ 127:112
    g1[4] = (int32_t)(nrows & 0xFFFFu);             // tile_dim1=nrows, tile_dim2=0
    g1[5] = W;                                      // tensor_dim0_stride lo32 = 33
    g1[6] = 0;                                      // stride0 hi16, stride1 lo16
    g1[7] = 0;                                      // stride1 hi32 (unused, 2D)
    // ---- D# groups 2/3: dims >=3 unused (tile_dim3/4 = 0) ----
    i32x4 g2 = {1, 1, 0, 0};                        // tensor_dim2=1, tensor_dim3=1
    i32x4 g3 = {0, 0, 0, 0};
    i32x8 gx = {0, 0, 0, 0, 0, 0, 0, 0};            // trailing group (zero-filled)
    if (is_store)
        __builtin_amdgcn_tensor_store_from_lds(g0, g1, g2, g3, gx, 0);
    else
        __builtin_amdgcn_tensor_load_to_lds(g0, g1, g2, g3, gx, 0);
}

__global__ __launch_bounds__(ROWS_PER_BLOCK)
void lpc_to_parcor_kernel(const float* __restrict__ a, float* __restrict__ out,
                          int rows) {
    __shared__ float tile[ROWS_PER_BLOCK * W];

    const int row0 = blockIdx.x * ROWS_PER_BLOCK;
    int rem = rows - row0;
    const uint32_t nrows = (uint32_t)(rem < ROWS_PER_BLOCK ? rem : ROWS_PER_BLOCK);
    const uint32_t lds_base = (uint32_t)(uintptr_t)&tile[0];

    // Wave 0 DMAs the 256x33 f32 tile into LDS (TENSORcnt is per-wave, so the
    // issuing wave waits, then the workgroup barrier publishes the tile).
    if (threadIdx.x < 32u) {
        tdm_issue(false, lds_base, (uint64_t)(uintptr_t)(a + (size_t)row0 * W), nrows);
        __builtin_amdgcn_s_wait_tensorcnt(0);
    }
    __syncthreads();

    const int t = (int)threadIdx.x;
    float* my = &tile[t * W];
    if (t < (int)rem) {
        // Pull the whole row into registers; all later indices are constants
        // after unrolling, so ar[] lives entirely in VGPRs (no scratch).
        float ar[ORDER];
#pragma unroll
        for (int j = 0; j < ORDER; ++j) ar[j] = my[j + 1] * GAMMA;
        // my[0] already holds K (gain) -- passes through unchanged.

        // Step-down recursion, in place. Pair (i, m-1-i) updated together:
        //   ar[i]     = (ar[i]     - km*ar[m-1-i]) * rcp(1-km^2)
        //   ar[m-1-i] = (ar[m-1-i] - km*ar[i]    ) * rcp(1-km^2)
        // v_rcp_f32 is a TRANS op: ~1 ulp, co-executes with the fma stream.
#pragma unroll
        for (int m = ORDER - 1; m >= 1; --m) {
            const float km = ar[m];
            my[m + 1] = km;                       // emit PARCOR k_m
            const float z  = 1.0f - km * km;
            const float rz = __builtin_amdgcn_rcpf(z);
#pragma unroll
            for (int i = 0; i < m / 2; ++i) {
                const float x = ar[i];
                const float y = ar[m - 1 - i];
                ar[i]         = (x - km * y) * rz;
                ar[m - 1 - i] = (y - km * x) * rz;
            }
            if (m & 1) {                          // middle element when m odd
                const float x = ar[(m - 1) / 2];
                ar[(m - 1) / 2] = (x - km * x) * rz;
            }
        }
        my[1] = ar[0];                            // k_0
    }
    __syncthreads();

    // Wave 0 DMAs the finished tile back out.
    if (threadIdx.x < 32u) {
        tdm_issue(true, lds_base, (uint64_t)(uintptr_t)(out + (size_t)row0 * W), nrows);
        __builtin_amdgcn_s_wait_tensorcnt(0);
    }
}

extern "C" void kernel_launch(void* const* d_in, const int* in_sizes, int n_in,
                              void* d_out, int out_size, void* d_ws, size_t ws_size,
                              hipStream_t stream) {
    const float* a = (const float*)d_in[0];
    float* out = (float*)d_out;
    const int rows = in_sizes[0] / W;                      // 2048*1024
    const int blocks = (rows + ROWS_PER_BLOCK - 1) / ROWS_PER_BLOCK;
    lpc_to_parcor_kernel<<<blocks, ROWS_PER_BLOCK, 0, stream>>>(a, out, rows);
}